// CoOp_34325378630026
// MI455X (gfx1250) — compile-verified
//
#include <hip/hip_runtime.h>
#include <stdint.h>

#define PLEN 2048
#define VOCAB 49408
#define DIM 512

#define MP 64            // rows per block
#define CC 128           // vocab columns per chunk
#define KT 128           // K per LDS stage
#define NCHUNK (VOCAB / CC)   // 386 (exact)
#define NSPLIT 8

typedef float v2f __attribute__((ext_vector_type(2)));
typedef float v8f __attribute__((ext_vector_type(8)));
typedef int   v4i __attribute__((vector_size(4 * sizeof(int))));

#define GLOBAL_AS __attribute__((address_space(1)))
#define LDS_AS    __attribute__((address_space(3)))

#if __has_builtin(__builtin_amdgcn_global_load_async_to_lds_b128)
#define HAVE_ASYNC_LDS 1
#else
#define HAVE_ASYNC_LDS 0
#endif

__device__ __forceinline__ void copy16_to_lds(const float* g, float* l) {
#if HAVE_ASYNC_LDS
    __builtin_amdgcn_global_load_async_to_lds_b128(
        (GLOBAL_AS v4i*)g, (LDS_AS v4i*)l, 0, 0);
#else
    *(float4*)l = *(const float4*)g;
#endif
}

__device__ __forceinline__ void async_wait_all() {
#if HAVE_ASYNC_LDS
#if __has_builtin(__builtin_amdgcn_s_wait_asynccnt)
    __builtin_amdgcn_s_wait_asynccnt(0);
#else
    asm volatile("s_wait_asynccnt 0" ::: "memory");
#endif
#endif
}

__global__ void init_res_kernel(unsigned long long* __restrict__ res) {
    int i = blockIdx.x * blockDim.x + threadIdx.x;
    if (i < PLEN) res[i] = 0xFFFFFFFFFFFFFFFFull;
}

// one wave per vocab row: c2[v] = sum_k clip[v][k]^2
__global__ void c2_kernel(const float* __restrict__ clip, float* __restrict__ c2) {
    int wave = threadIdx.x >> 5;
    int lane = threadIdx.x & 31;
    int v = blockIdx.x * 8 + wave;              // gridDim.x = VOCAB/8 (exact)
    const float4* row = (const float4*)(clip + (size_t)v * DIM);
    float s = 0.f;
#pragma unroll
    for (int j = 0; j < 4; ++j) {
        float4 q = row[lane + j * 32];
        s += q.x * q.x + q.y * q.y + q.z * q.z + q.w * q.w;
    }
#pragma unroll
    for (int off = 16; off; off >>= 1) s += __shfl_xor(s, off, 32);
    if (lane == 0) c2[v] = s;
}

__global__ __launch_bounds__(256) void argmin_kernel(
    const float* __restrict__ prompt, const float* __restrict__ clip,
    const float* __restrict__ c2, unsigned long long* __restrict__ res) {

    __shared__ float As[MP][DIM + 4];      // 64 x 516 f32 = 132 KB
    __shared__ float Bs[2][CC][KT + 4];    // 2 x 128 x 132 f32 = 135 KB (double buffer)

    const int tid  = threadIdx.x;
    const int lane = tid & 31;
    const int wave = tid >> 5;
    const int l15  = lane & 15;
    const int hi   = lane >> 4;

    const int rowBase = blockIdx.y * MP;

    // Issue one B stage (CC cols x KT k-values) through the async path.
    auto issueB = [&](int buf, int c, int k0) {
        const size_t vb = (size_t)c * CC;
#pragma unroll
        for (int i = 0; i < (CC * KT / 4) / 256; ++i) {   // 16 x b128 per thread
            int idx = tid + i * 256;
            int col = idx >> 5;        // KT/4 = 32 float4 per column
            int kf  = idx & 31;
            copy16_to_lds(clip + (vb + col) * (size_t)DIM + k0 + kf * 4,
                          &Bs[buf][col][kf * 4]);
        }
    };

    // Stage the A panel (MP x DIM) through the async path as well; it is
    // covered by the first s_wait_asynccnt(0) + barrier below.
    {
        const float* src = prompt + (size_t)rowBase * DIM;
#pragma unroll
        for (int i = 0; i < (MP * DIM / 4) / 256; ++i) {  // 32 x b128 per thread
            int idx = tid + i * 256;
            int r   = idx >> 7;        // DIM/4 = 128 float4 per row
            int kf  = idx & 127;
            copy16_to_lds(src + (size_t)r * DIM + kf * 4, &As[r][kf * 4]);
        }
    }

    // wave -> 16x16 tiles: 4 row-tiles x 8 col-tiles per chunk; each wave owns
    // one row-tile and 4 col-tiles (4 independent WMMA chains for ILP).
    const int rowTile  = wave >> 1;     // 0..3
    const int colGroup = wave & 1;      // 0..1
    const int rBase    = rowTile * 16;

    const float* aRow = &As[rBase + l15][0];
    int bOff[4];
#pragma unroll
    for (int j = 0; j < 4; ++j)
        bOff[j] = (colGroup * 64 + j * 16 + l15) * (KT + 4) + 2 * hi;
    const float* bBase0 = &Bs[0][0][0];
    const float* bBase1 = &Bs[1][0][0];

    unsigned long long best[8];
#pragma unroll
    for (int i = 0; i < 8; ++i) best[i] = 0xFFFFFFFFFFFFFFFFull;

    const v8f vzero = {};

    // Software pipeline: prefetch first stage, then wait/barrier/issue/compute.
    int buf = 0;
    issueB(0, blockIdx.x, 0);

    for (int c = blockIdx.x; c < NCHUNK; c += gridDim.x) {
        const size_t vBase = (size_t)c * CC;

        v8f acc[4];
#pragma unroll
        for (int j = 0; j < 4; ++j) acc[j] = vzero;

        for (int k0 = 0; k0 < DIM; k0 += KT) {
            async_wait_all();      // this wave's portion of stage (c,k0) landed
            __syncthreads();       // all portions landed; prior compute retired

            // Prefetch the next stage into the other buffer (safe: the barrier
            // above proved every wave finished reading it).
            int cN  = c;
            int k0N = k0 + KT;
            if (k0N == DIM) { cN = c + gridDim.x; k0N = 0; }
            if (cN < NCHUNK) issueB(buf ^ 1, cN, k0N);

            const float* bb = buf ? bBase1 : bBase0;
#pragma unroll
            for (int kk = 0; kk < KT; kk += 4) {
                // A 16x4 layout: lane l, vgpr i -> M=l&15, K = i + 2*(l>>4)
                v2f a = *(const v2f*)(aRow + k0 + kk + 2 * hi);
#pragma unroll
                for (int j = 0; j < 4; ++j) {
                    v2f b = *(const v2f*)(bb + bOff[j] + kk);
                    acc[j] = __builtin_amdgcn_wmma_f32_16x16x4_f32(
                        false, a, false, b, (short)0, acc[j], false, false);
                }
            }
            buf ^= 1;
        }

        // score = c2[v] - 2*dot ; monotone-ordered key packed with column id.
        // C/D layout: vgpr i, lane l -> row = i + 8*(l>>4), col = l&15.
#pragma unroll
        for (int j = 0; j < 4; ++j) {
            int vcol = (int)vBase + colGroup * 64 + j * 16 + l15;
            float cv = c2[vcol];
#pragma unroll
            for (int i = 0; i < 8; ++i) {
                float s = cv - 2.0f * acc[j][i];
                unsigned ku = __float_as_uint(s);
                ku = (ku & 0x80000000u) ? ~ku : (ku | 0x80000000u);
                unsigned long long pk =
                    ((unsigned long long)ku << 32) | (unsigned)vcol;
                if (pk < best[i]) best[i] = pk;
            }
        }
    }

    // Reduce over the 16 column-lanes of each half-wave, then merge globally.
#pragma unroll
    for (int i = 0; i < 8; ++i) {
        unsigned long long b = best[i];
#pragma unroll
        for (int off = 1; off <= 8; off <<= 1) {
            unsigned lo = (unsigned)b;
            unsigned hh = (unsigned)(b >> 32);
            lo = __shfl_xor(lo, off, 32);
            hh = __shfl_xor(hh, off, 32);
            unsigned long long o = ((unsigned long long)hh << 32) | lo;
            if (o < b) b = o;
        }
        if (l15 == 0) {
            atomicMin(&res[rowBase + rBase + i + 8 * hi], b);
        }
    }
}

__global__ void finalize_kernel(const float* __restrict__ prompt,
                                const unsigned long long* __restrict__ res,
                                float* __restrict__ out) {
    int i = blockIdx.x * blockDim.x + threadIdx.x;
    const int n = PLEN * DIM;
    if (i < n) {
        out[i] = prompt[i];
    } else if (i < n + PLEN) {
        out[i] = (float)(unsigned)(res[i - n] & 0xFFFFFFFFull);
    }
}

extern "C" void kernel_launch(void* const* d_in, const int* in_sizes, int n_in,
                              void* d_out, int out_size, void* d_ws, size_t ws_size,
                              hipStream_t stream) {
    const float* prompt = (const float*)d_in[0];
    const float* clip   = (const float*)d_in[1];

    unsigned long long* res = (unsigned long long*)d_ws;                       // 16 KB
    float* c2 = (float*)((char*)d_ws + PLEN * sizeof(unsigned long long));     // 198 KB

    float* out = (float*)d_out;

    init_res_kernel<<<dim3((PLEN + 255) / 256), dim3(256), 0, stream>>>(res);
    c2_kernel<<<dim3(VOCAB / 8), dim3(256), 0, stream>>>(clip, c2);
    argmin_kernel<<<dim3(NSPLIT, PLEN / MP), dim3(256), 0, stream>>>(prompt, clip, c2, res);
    finalize_kernel<<<dim3((PLEN * DIM + PLEN + 255) / 256), dim3(256), 0, stream>>>(prompt, res, out);
}